// LinkPredModel_16947940950167
// MI455X (gfx1250) — compile-verified
//
#include <hip/hip_runtime.h>
#include <hip/hip_bf16.h>
#include <math.h>

// ---------------------------------------------------------------------------
// LinkPred GNN: 2x GATv2 (WMMA fp32 GEMMs + atomic segment softmax) + MLP head
// Target: gfx1250 (MI455X), wave32. fp32 throughout (matches reference dtype;
// GEMMs are ~7.4 GFLOP total -> far from any compute ceiling, and the whole
// ~117MB workspace is L2-resident on the 192MB L2, so the edge scatter/gather
// passes are L2-BW bound).
// ---------------------------------------------------------------------------

typedef float v2f __attribute__((ext_vector_type(2)));
typedef float v8f __attribute__((ext_vector_type(8)));

__device__ __forceinline__ void atomic_max_f32(float* addr, float v) {
  // sign-punned float max: int-max for non-negative, uint-min for negative.
  if (v >= 0.f) atomicMax((int*)addr, __float_as_int(v));
  else          atomicMin((unsigned int*)addr, __float_as_uint(v));
}

__global__ void fill_f32(float* __restrict__ p, float v, int n) {
  int i = blockIdx.x * blockDim.x + threadIdx.x;
  if (i < n) p[i] = v;
}

// ---------------------------------------------------------------------------
// C[M,NC] = X[M,K] @ W[K,NC] + bias[NC]   (row-major, fp32, WMMA 16x16x4)
// One wave computes a 16-row strip; A fragments held in regs across N tiles.
// ---------------------------------------------------------------------------
template<int K, int NC>
__global__ __launch_bounds__(256)
void gemm_wmma(const float* __restrict__ X, const float* __restrict__ W,
               const float* __restrict__ bias, float* __restrict__ C, int M)
{
  __shared__ float sW[K * NC];
  for (int i = threadIdx.x; i < K * NC; i += blockDim.x) sW[i] = W[i];
  __syncthreads();

  const int wave = threadIdx.x >> 5;
  const int lane = threadIdx.x & 31;
  const int m0 = (blockIdx.x * 8 + wave) * 16;
  if (m0 >= M) return;

  const int l16   = lane & 15;
  const int khalf = (lane >> 4) * 2;          // lanes 0-15: K{0,1}; 16-31: K{2,3}
  const int rowA  = m0 + l16;
  const bool rowOK = rowA < M;
  const float* xrow = X + (size_t)rowA * K;

  v2f afrag[K / 4];
#pragma unroll
  for (int kt = 0; kt < K / 4; ++kt) {
    const int kb = kt * 4 + khalf;
    afrag[kt].x = rowOK ? xrow[kb]     : 0.f;
    afrag[kt].y = rowOK ? xrow[kb + 1] : 0.f;
  }

#pragma unroll
  for (int n0 = 0; n0 < NC; n0 += 16) {
    v8f acc = {0.f, 0.f, 0.f, 0.f, 0.f, 0.f, 0.f, 0.f};
#pragma unroll
    for (int kt = 0; kt < K / 4; ++kt) {
      const int kb = kt * 4 + khalf;
      v2f b;
      b.x = sW[(kb)     * NC + n0 + l16];
      b.y = sW[(kb + 1) * NC + n0 + l16];
      acc = __builtin_amdgcn_wmma_f32_16x16x4_f32(
          false, afrag[kt], false, b, (short)0, acc, false, false);
    }
    const int rbase = m0 + ((lane >> 4) << 3);  // lanes 16-31 hold rows +8
    const float bv = bias ? bias[n0 + l16] : 0.f;
#pragma unroll
    for (int r = 0; r < 8; ++r) {
      const int row = rbase + r;
      if (row < M) C[(size_t)row * NC + n0 + l16] = acc[r] + bv;
    }
  }
}

// ---------------------------------------------------------------------------
// GATv2 edge passes. Edges 0..E-1 come from edge_index, e in [E, E+N) are the
// implicit self-loops. C (per-head width) is fixed to 64.
// ---------------------------------------------------------------------------
__device__ __forceinline__ void edge_sd(const int* __restrict__ ei, int E,
                                        int e, int& s, int& d) {
  if (e < E) { s = ei[e]; d = ei[E + e]; }
  else       { s = d = e - E; }
}

template<int H>
__global__ __launch_bounds__(256)
void gat_logits(const int* __restrict__ ei, int E, int N,
                const float* __restrict__ xl, const float* __restrict__ xr,
                const float* __restrict__ att,       // [H*64]
                float* __restrict__ mmax,            // [N*H], pre-init -inf
                float* __restrict__ elog)            // [(E+N)*H]
{
  const int e = blockIdx.x * 8 + (threadIdx.x >> 5);
  const int EN = E + N;
  if (e >= EN) return;
  const int lane = threadIdx.x & 31;
  int s, d; edge_sd(ei, E, e, s, d);

  for (int h = 0; h < H; ++h) {
    const float2 a = ((const float2*)(xl + ((size_t)s * H + h) * 64))[lane];
    const float2 b = ((const float2*)(xr + ((size_t)d * H + h) * 64))[lane];
    const float2 w = ((const float2*)(att + h * 64))[lane];
    float v0 = a.x + b.x; v0 = v0 > 0.f ? v0 : 0.2f * v0;
    float v1 = a.y + b.y; v1 = v1 > 0.f ? v1 : 0.2f * v1;
    float sum = v0 * w.x + v1 * w.y;
#pragma unroll
    for (int off = 16; off; off >>= 1) sum += __shfl_xor(sum, off, 32);
    if (lane == 0) {
      elog[(size_t)e * H + h] = sum;
      atomic_max_f32(&mmax[d * H + h], sum);
    }
  }
}

template<int H>
__global__ __launch_bounds__(256)
void gat_exp(const int* __restrict__ ei, int E, int N,
             const float* __restrict__ mmax, float* __restrict__ dsum,
             float* __restrict__ elog)
{
  const int EN = E + N;
  const int i = blockIdx.x * blockDim.x + threadIdx.x;
  if (i >= EN * H) return;
  const int e = i / H;
  const int h = i - e * H;
  int s, d; edge_sd(ei, E, e, s, d); (void)s;
  const float ev = __expf(elog[i] - mmax[d * H + h]);
  elog[i] = ev;
  atomicAdd(&dsum[d * H + h], ev);
}

template<int H>
__global__ __launch_bounds__(256)
void gat_agg(const int* __restrict__ ei, int E, int N,
             const float* __restrict__ xl, const float* __restrict__ dsum,
             const float* __restrict__ elog,
             float* __restrict__ hout)    // pre-init to x@Wres + bias
{
  const int e = blockIdx.x * 8 + (threadIdx.x >> 5);
  const int EN = E + N;
  if (e >= EN) return;
  const int lane = threadIdx.x & 31;
  int s, d; edge_sd(ei, E, e, s, d);

  for (int h = 0; h < H; ++h) {
    const float alpha = elog[(size_t)e * H + h] / (dsum[d * H + h] + 1e-16f);
    const float2 v = ((const float2*)(xl + ((size_t)s * H + h) * 64))[lane];
    float* dp = hout + ((size_t)d * H + h) * 64 + 2 * lane;
    atomicAdd(dp,     alpha * v.x);
    atomicAdd(dp + 1, alpha * v.y);
  }
}

// ---------------------------------------------------------------------------
// MLP head.
// ---------------------------------------------------------------------------
__global__ __launch_bounds__(256)
void head_feats(const int* __restrict__ lei, int L,
                const float* __restrict__ h2, const float* __restrict__ ef,
                float* __restrict__ feats, float* __restrict__ stats0 /*[160]*/)
{
  __shared__ float ssum[80], ssq[80];
  for (int i = threadIdx.x; i < 80; i += blockDim.x) { ssum[i] = 0.f; ssq[i] = 0.f; }
  __syncthreads();
  const int wave = threadIdx.x >> 5, lane = threadIdx.x & 31;
  const int r = blockIdx.x * 8 + wave;
  if (r < L) {
    const int s = lei[r], d = lei[L + r];
    for (int j = lane; j < 80; j += 32) {
      const float f = (j < 64) ? h2[(size_t)s * 64 + j] * h2[(size_t)d * 64 + j]
                               : ef[(size_t)r * 16 + (j - 64)];
      feats[(size_t)r * 80 + j] = f;
      atomicAdd(&ssum[j], f);
      atomicAdd(&ssq[j], f * f);
    }
  }
  __syncthreads();
  for (int i = threadIdx.x; i < 80; i += blockDim.x) {
    atomicAdd(&stats0[i],      ssum[i]);
    atomicAdd(&stats0[80 + i], ssq[i]);
  }
}

__global__ void bn_fold(const float* __restrict__ stats,
                        const float* __restrict__ g, const float* __restrict__ b,
                        float* __restrict__ scale, float* __restrict__ shift,
                        int D, float invL)
{
  const int i = blockIdx.x * blockDim.x + threadIdx.x;
  if (i < D) {
    const float mean = stats[i] * invL;
    const float var  = stats[D + i] * invL - mean * mean;
    const float sc   = g[i] * rsqrtf(var + 1e-5f);
    scale[i] = sc;
    shift[i] = b[i] - mean * sc;
  }
}

__global__ __launch_bounds__(256)
void head_lin0(const float* __restrict__ feats,
               const float* __restrict__ scale0, const float* __restrict__ shift0,
               const float* __restrict__ Win, const float* __restrict__ bin, int L,
               float* __restrict__ o1, float* __restrict__ stats1 /*[32]*/)
{
  __shared__ float sW[80 * 16], sSc[80], sSh[80], sB[16], ssum[16], ssq[16];
  for (int i = threadIdx.x; i < 80 * 16; i += blockDim.x) sW[i] = Win[i];
  for (int i = threadIdx.x; i < 80; i += blockDim.x) { sSc[i] = scale0[i]; sSh[i] = shift0[i]; }
  for (int i = threadIdx.x; i < 16; i += blockDim.x) { sB[i] = bin[i]; ssum[i] = 0.f; ssq[i] = 0.f; }
  __syncthreads();
  const int r = blockIdx.x * blockDim.x + threadIdx.x;
  if (r < L) {
    float acc[16];
#pragma unroll
    for (int k = 0; k < 16; ++k) acc[k] = sB[k];
    for (int j = 0; j < 80; ++j) {
      const float xn = feats[(size_t)r * 80 + j] * sSc[j] + sSh[j];
#pragma unroll
      for (int k = 0; k < 16; ++k) acc[k] += xn * sW[j * 16 + k];
    }
#pragma unroll
    for (int k = 0; k < 16; ++k) {
      o1[(size_t)r * 16 + k] = acc[k];
      atomicAdd(&ssum[k], acc[k]);
      atomicAdd(&ssq[k],  acc[k] * acc[k]);
    }
  }
  __syncthreads();
  if (threadIdx.x < 16) {
    atomicAdd(&stats1[threadIdx.x],      ssum[threadIdx.x]);
    atomicAdd(&stats1[16 + threadIdx.x], ssq[threadIdx.x]);
  }
}

__global__ __launch_bounds__(256)
void head_lin1(const float* __restrict__ o1,
               const float* __restrict__ scale1, const float* __restrict__ shift1,
               const float* __restrict__ Whid, const float* __restrict__ bhid, int L,
               float* __restrict__ o2, float* __restrict__ stats2 /*[8]*/)
{
  __shared__ float sW[64], sSc[16], sSh[16], sB[4], ssum[4], ssq[4];
  for (int i = threadIdx.x; i < 64; i += blockDim.x) sW[i] = Whid[i];
  for (int i = threadIdx.x; i < 16; i += blockDim.x) { sSc[i] = scale1[i]; sSh[i] = shift1[i]; }
  for (int i = threadIdx.x; i < 4; i += blockDim.x) { sB[i] = bhid[i]; ssum[i] = 0.f; ssq[i] = 0.f; }
  __syncthreads();
  const int r = blockIdx.x * blockDim.x + threadIdx.x;
  if (r < L) {
    float acc[4];
#pragma unroll
    for (int k = 0; k < 4; ++k) acc[k] = sB[k];
#pragma unroll
    for (int j = 0; j < 16; ++j) {
      float xn = o1[(size_t)r * 16 + j] * sSc[j] + sSh[j];
      xn = xn > 0.f ? xn : 0.01f * xn;               // LeakyReLU(0.01)
#pragma unroll
      for (int k = 0; k < 4; ++k) acc[k] += xn * sW[j * 4 + k];
    }
#pragma unroll
    for (int k = 0; k < 4; ++k) {
      o2[(size_t)r * 4 + k] = acc[k];
      atomicAdd(&ssum[k], acc[k]);
      atomicAdd(&ssq[k],  acc[k] * acc[k]);
    }
  }
  __syncthreads();
  if (threadIdx.x < 4) {
    atomicAdd(&stats2[threadIdx.x],     ssum[threadIdx.x]);
    atomicAdd(&stats2[4 + threadIdx.x], ssq[threadIdx.x]);
  }
}

__global__ __launch_bounds__(256)
void head_lin2(const float* __restrict__ o2,
               const float* __restrict__ scale2, const float* __restrict__ shift2,
               const float* __restrict__ Wout, const float* __restrict__ bout,
               int L, float* __restrict__ out)
{
  const int r = blockIdx.x * blockDim.x + threadIdx.x;
  if (r >= L) return;
  float acc = bout[0];
#pragma unroll
  for (int j = 0; j < 4; ++j) {
    float xn = o2[(size_t)r * 4 + j] * scale2[j] + shift2[j];
    xn = xn > 0.f ? xn : 0.01f * xn;
    acc += xn * Wout[j];
  }
  out[r] = acc;
}

// ---------------------------------------------------------------------------
extern "C" void kernel_launch(void* const* d_in, const int* in_sizes, int n_in,
                              void* d_out, int out_size, void* d_ws, size_t ws_size,
                              hipStream_t stream)
{
  const float* x    = (const float*)d_in[0];
  const int*   ei   = (const int*)d_in[1];
  const int*   lei  = (const int*)d_in[2];
  const float* ef   = (const float*)d_in[3];
  const float* W1l  = (const float*)d_in[4];
  const float* b1l  = (const float*)d_in[5];
  const float* W1r  = (const float*)d_in[6];
  const float* b1r  = (const float*)d_in[7];
  const float* att1 = (const float*)d_in[8];
  const float* Wres1= (const float*)d_in[9];
  const float* bias1= (const float*)d_in[10];
  const float* W2l  = (const float*)d_in[11];
  const float* b2l  = (const float*)d_in[12];
  const float* W2r  = (const float*)d_in[13];
  const float* b2r  = (const float*)d_in[14];
  const float* att2 = (const float*)d_in[15];
  const float* Wres2= (const float*)d_in[16];
  const float* bias2= (const float*)d_in[17];
  const float* bn0g = (const float*)d_in[18];
  const float* bn0b = (const float*)d_in[19];
  const float* Win  = (const float*)d_in[20];
  const float* bin_ = (const float*)d_in[21];
  const float* bn1g = (const float*)d_in[22];
  const float* bn1b = (const float*)d_in[23];
  const float* Whid = (const float*)d_in[24];
  const float* bhid = (const float*)d_in[25];
  const float* bn2g = (const float*)d_in[26];
  const float* bn2b = (const float*)d_in[27];
  const float* Wout = (const float*)d_in[28];
  const float* bout = (const float*)d_in[29];
  float* out = (float*)d_out;

  const int N  = in_sizes[0] / 128;
  const int E  = in_sizes[1] / 2;
  const int L  = in_sizes[2] / 2;
  const int EN = E + N;

  // ---- workspace layout (floats); peak ~117 MB, deliberately < 192MB L2 ----
  float* ws = (float*)d_ws;
  size_t o = 0;
  float* xl1 = ws + o; o += (size_t)N * 128;
  float* xr1 = ws + o; o += (size_t)N * 128;
  float* h1  = ws + o; o += (size_t)N * 128;   // res1+bias1, then aggregated
  float* m1  = ws + o; o += (size_t)N * 2;
  float* d1  = ws + o; o += (size_t)N * 2;
  float* el1 = ws + o; o += (size_t)EN * 2;
  float* h2  = ws + o; o += (size_t)N * 64;    // res2+bias2, then aggregated
  float* m2  = ws + o; o += (size_t)N;
  float* d2  = ws + o; o += (size_t)N;
  float* el2 = ws + o; o += (size_t)EN;
  float* o1  = ws + o; o += (size_t)L * 16;
  float* o2  = ws + o; o += (size_t)L * 4;
  float* stats0 = ws + o; o += 160;
  float* stats1 = ws + o; o += 32;
  float* stats2 = ws + o; o += 8;
  float* scale0 = ws + o; o += 80;
  float* shift0 = ws + o; o += 80;
  float* scale1 = ws + o; o += 16;
  float* shift1 = ws + o; o += 16;
  float* scale2 = ws + o; o += 4;
  float* shift2 = ws + o; o += 4;
  // Reused regions (dead by the time they are overwritten):
  float* xl2   = xl1;                      // N*64  (layer1 xl dead)
  float* xr2   = xl1 + (size_t)N * 64;     // N*64
  float* feats = xl1;                      // L*80 <= 3*N*128 region
  (void)ws_size; (void)n_in; (void)out_size;

  auto cdiv = [](long long a, long long b) { return (unsigned)((a + b - 1) / b); };

  // ---------------- Layer 1 (H=2, C=64) ----------------
  gemm_wmma<128, 128><<<cdiv(N, 128), 256, 0, stream>>>(x, W1l,  b1l,   xl1, N);
  gemm_wmma<128, 128><<<cdiv(N, 128), 256, 0, stream>>>(x, W1r,  b1r,   xr1, N);
  gemm_wmma<128, 128><<<cdiv(N, 128), 256, 0, stream>>>(x, Wres1, bias1, h1, N);
  fill_f32<<<cdiv(N * 2, 256), 256, 0, stream>>>(m1, -INFINITY, N * 2);
  fill_f32<<<cdiv(N * 2, 256), 256, 0, stream>>>(d1, 0.f, N * 2);
  gat_logits<2><<<cdiv(EN, 8), 256, 0, stream>>>(ei, E, N, xl1, xr1, att1, m1, el1);
  gat_exp<2><<<cdiv((long long)EN * 2, 256), 256, 0, stream>>>(ei, E, N, m1, d1, el1);
  gat_agg<2><<<cdiv(EN, 8), 256, 0, stream>>>(ei, E, N, xl1, d1, el1, h1);

  // ---------------- Layer 2 (H=1, C=64) ----------------
  gemm_wmma<128, 64><<<cdiv(N, 128), 256, 0, stream>>>(h1, W2l,  b2l,   xl2, N);
  gemm_wmma<128, 64><<<cdiv(N, 128), 256, 0, stream>>>(h1, W2r,  b2r,   xr2, N);
  gemm_wmma<128, 64><<<cdiv(N, 128), 256, 0, stream>>>(h1, Wres2, bias2, h2, N);
  fill_f32<<<cdiv(N, 256), 256, 0, stream>>>(m2, -INFINITY, N);
  fill_f32<<<cdiv(N, 256), 256, 0, stream>>>(d2, 0.f, N);
  gat_logits<1><<<cdiv(EN, 8), 256, 0, stream>>>(ei, E, N, xl2, xr2, att2, m2, el2);
  gat_exp<1><<<cdiv(EN, 256), 256, 0, stream>>>(ei, E, N, m2, d2, el2);
  gat_agg<1><<<cdiv(EN, 8), 256, 0, stream>>>(ei, E, N, xl2, d2, el2, h2);

  // ---------------- MLP head over label edges ----------------
  fill_f32<<<1, 256, 0, stream>>>(stats0, 0.f, 160 + 32 + 8);  // stats are contiguous
  head_feats<<<cdiv(L, 8), 256, 0, stream>>>(lei, L, h2, ef, feats, stats0);
  bn_fold<<<1, 128, 0, stream>>>(stats0, bn0g, bn0b, scale0, shift0, 80, 1.f / (float)L);
  head_lin0<<<cdiv(L, 256), 256, 0, stream>>>(feats, scale0, shift0, Win, bin_, L, o1, stats1);
  bn_fold<<<1, 32, 0, stream>>>(stats1, bn1g, bn1b, scale1, shift1, 16, 1.f / (float)L);
  head_lin1<<<cdiv(L, 256), 256, 0, stream>>>(o1, scale1, shift1, Whid, bhid, L, o2, stats2);
  bn_fold<<<1, 32, 0, stream>>>(stats2, bn2g, bn2b, scale2, shift2, 4, 1.f / (float)L);
  head_lin2<<<cdiv(L, 256), 256, 0, stream>>>(o2, scale2, shift2, Wout, bout, L, out);
}